// MultiHeadDotProduct_67946382623506
// MI455X (gfx1250) — compile-verified
//
#include <hip/hip_runtime.h>
#include <hip/hip_bf16.h>

// CDNA5 (gfx1250) dense multi-head attention, fp32 end-to-end via
// V_WMMA_F32_16X16X4_F32 (full-precision matrix pipe).
//
// Problem: BS=512, EMBED=256, NHEAD=8, HDIM=32; edge list is the full 512x512
// clique, so the reference is exactly dense softmax attention + projections.

typedef float v2f __attribute__((ext_vector_type(2)));
typedef float v8f __attribute__((ext_vector_type(8)));

#define BS    512
#define EMBED 256
#define NHEAD 8
#define HDIM  32
// LDS row stride (floats) for the 16x512 score tile: 516 -> bank stride 4,
// conflict-free b64 access pattern, and 8-byte aligned rows (516*4 % 8 == 0).
#define LDSW  516

__device__ __forceinline__ v8f wmma_f32(v2f a, v2f b, v8f c) {
  // 8-arg form: (neg_a, A, neg_b, B, c_mod, C, reuse_a, reuse_b)
  return __builtin_amdgcn_wmma_f32_16x16x4_f32(false, a, false, b, (short)0, c,
                                               false, false);
}

// out[m,n] = scale * (sum_k X[m,k] * W[n,k] + bias[n])
// One wave per 16x16 output tile; blockDim must be a multiple of 32.
__global__ void gemm_xt_bias(const float* __restrict__ X,
                             const float* __restrict__ W,
                             const float* __restrict__ bias,
                             float* __restrict__ out,
                             int M, int N, int K, float scale) {
  const int lane = threadIdx.x & 31;
  const int wave = threadIdx.x >> 5;
  const int tile = blockIdx.x * (blockDim.x >> 5) + wave;
  const int ntn  = N >> 4;
  const int tm   = tile / ntn;
  const int tn   = tile - tm * ntn;
  if (tm * 16 >= M) return;              // wave-uniform: EXEC stays all-ones
  const int lh  = lane >> 4;             // lane half (0/1)
  const int l16 = lane & 15;
  const int arow = tm * 16 + l16;        // A: row M = lane&15
  const int bcol = tn * 16 + l16;        // B/C/D: col N = lane&15

  v8f acc = {};
  for (int k = 0; k < K; k += 4) {
    const int kk = k + lh * 2;           // VGPR0/1 hold K = 2h, 2h+1
    v2f a = *(const v2f*)(X + arow * K + kk);
    v2f b = *(const v2f*)(W + bcol * K + kk);   // B[k][n] = W[n][k]
    acc = wmma_f32(a, b, acc);
  }
  const float bn = bias[bcol];
#pragma unroll
  for (int v = 0; v < 8; ++v) {          // D: row = v + 8*half, col = bcol
    const int row = tm * 16 + v + lh * 8;
    out[row * N + bcol] = scale * (acc[v] + bn);
  }
}

// One wave per (head, 16-query tile). Q already carries the 1/sqrt(HDIM) scale.
__global__ void attn_head_tile(const float* __restrict__ Q,
                               const float* __restrict__ Km,
                               const float* __restrict__ V,
                               float* __restrict__ agg) {
  __shared__ float S[16 * LDSW];
  const int lane = threadIdx.x & 31;
  const int lh   = lane >> 4;
  const int l16  = lane & 15;
  const int h    = blockIdx.x & (NHEAD - 1);
  const int tm   = blockIdx.x >> 3;      // query tile 0..31
  const int m0   = tm * 16;

  const float* Qh = Q  + h * HDIM;
  const float* Kh = Km + h * HDIM;
  const float* Vh = V  + h * HDIM;

  // ---- S = Qh(16x32) @ Kh^T -> 32 column tiles of 16, staged in LDS ----
  const int qrow = m0 + l16;
  for (int tn = 0; tn < BS / 16; ++tn) {
    const int krow = tn * 16 + l16;
    v8f acc = {};
#pragma unroll
    for (int k = 0; k < HDIM; k += 4) {
      const int kk = k + lh * 2;
      v2f a = *(const v2f*)(Qh + qrow * EMBED + kk);
      v2f b = *(const v2f*)(Kh + krow * EMBED + kk);  // B[k][n] = Kh[n][k]
      acc = wmma_f32(a, b, acc);
    }
#pragma unroll
    for (int v = 0; v < 8; ++v)
      S[(v + lh * 8) * LDSW + tn * 16 + l16] = acc[v];
  }
  __syncthreads();

  // ---- row softmax over 512 keys: lane pair (r, r+16) splits the row ----
  {
    const int r = l16;
    float* row = &S[r * LDSW + lh * 256];
    float mx = -__builtin_inff();
    for (int c = 0; c < 256; ++c) mx = fmaxf(mx, row[c]);
    mx = fmaxf(mx, __shfl_xor(mx, 16));
    float sum = 0.0f;
    for (int c = 0; c < 256; ++c) {
      const float e = __expf(row[c] - mx);
      row[c] = e;
      sum += e;
    }
    sum += __shfl_xor(sum, 16);
    const float inv = 1.0f / sum;
    for (int c = 0; c < 256; ++c) row[c] *= inv;
  }
  __syncthreads();

  // ---- O = P(16x512) @ Vh(512x32): K=512, two N-tiles of 16 ----
  for (int tn = 0; tn < HDIM / 16; ++tn) {
    v8f acc = {};
    for (int k = 0; k < BS; k += 4) {
      const int kk = k + lh * 2;
      v2f a = *(const v2f*)(&S[l16 * LDSW + kk]);   // conflict-free ds_load_b64
      v2f b;
      b.x = Vh[(kk + 0) * EMBED + tn * 16 + l16];   // B[k][n] = Vh[k][n]
      b.y = Vh[(kk + 1) * EMBED + tn * 16 + l16];
      acc = wmma_f32(a, b, acc);
    }
#pragma unroll
    for (int v = 0; v < 8; ++v)
      agg[(m0 + v + lh * 8) * EMBED + h * HDIM + tn * 16 + l16] = acc[v];
  }
}

extern "C" void kernel_launch(void* const* d_in, const int* in_sizes, int n_in,
                              void* d_out, int out_size, void* d_ws, size_t ws_size,
                              hipStream_t stream) {
  (void)in_sizes; (void)n_in; (void)out_size; (void)ws_size;
  const float* feats = (const float*)d_in[0];
  // d_in[1] = edge_index, d_in[2] = edge_attr: full clique by construction -> dense attention.
  const float* Wq = (const float*)d_in[3];
  const float* bq = (const float*)d_in[4];
  const float* Wk = (const float*)d_in[5];
  const float* bk = (const float*)d_in[6];
  const float* Wv = (const float*)d_in[7];
  const float* bv = (const float*)d_in[8];
  const float* Wo = (const float*)d_in[9];
  const float* bo = (const float*)d_in[10];
  float* out = (float*)d_out;

  float* q  = (float*)d_ws;
  float* k  = q  + BS * EMBED;
  float* v  = k  + BS * EMBED;
  float* ag = v  + BS * EMBED;

  const float scale = 0.17677669529663687f;  // 1/sqrt(HDIM)

  const int tiles   = (BS / 16) * (EMBED / 16);  // 512 tiles
  const int threads = 128;                       // 4 waves/block, 1 tile/wave
  const int blocks  = tiles / (threads / 32);    // 128 blocks

  gemm_xt_bias<<<blocks, threads, 0, stream>>>(feats, Wq, bq, q, BS, EMBED, EMBED, scale);
  gemm_xt_bias<<<blocks, threads, 0, stream>>>(feats, Wk, bk, k, BS, EMBED, EMBED, 1.0f);
  gemm_xt_bias<<<blocks, threads, 0, stream>>>(feats, Wv, bv, v, BS, EMBED, EMBED, 1.0f);

  attn_head_tile<<<NHEAD * (BS / 16), 32, 0, stream>>>(q, k, v, ag);

  gemm_xt_bias<<<blocks, threads, 0, stream>>>(ag, Wo, bo, out, BS, EMBED, EMBED, 1.0f);
}